// GCNNet_40535901339683
// MI455X (gfx1250) — compile-verified
//
#include <hip/hip_runtime.h>
#include <math.h>

#define BN_EPS 1e-5f

typedef __attribute__((ext_vector_type(2))) float v2f;
typedef __attribute__((ext_vector_type(8))) float v8f;

// ---------------- degree / norm ----------------

__global__ void deg_init_kernel(float* __restrict__ deg, int n) {
  int i = blockIdx.x * blockDim.x + threadIdx.x;
  if (i < n) deg[i] = 1.0f;  // self loop contributes 1 to every node's degree
}

__global__ void deg_accum_kernel(const int* __restrict__ dst, float* __restrict__ deg, int e) {
  int i = blockIdx.x * blockDim.x + threadIdx.x;
  if (i < e) atomicAdd(&deg[dst[i]], 1.0f);
}

__global__ void dinv_kernel(float* __restrict__ deg, int n) {
  int i = blockIdx.x * blockDim.x + threadIdx.x;
  if (i < n) {
    float d = deg[i];
    deg[i] = (d > 0.0f) ? rsqrtf(d) : 0.0f;  // in-place: deg buffer becomes dinv
  }
}

// ---------------- fp32 WMMA GEMM: Y[N,DOUT] = X[N,DIN] @ W[DIN,DOUT] ----------------
// One wave per 16x16 output tile; block = (32, DOUT/16) waves sharing a 16xDIN
// X stripe staged in padded LDS. K-loop uses V_WMMA_F32_16X16X4_F32.
// A(16x4) lane layout: lanes 0-15 -> M=lane, K={k,k+1}; lanes 16-31 -> M=lane-16, K={k+2,k+3}.
// B(4x16) lane layout mirrors A; C/D(16x16): VGPR g, lanes 0-15 -> M=g, lanes 16-31 -> M=g+8.

template <int DIN, int DOUT>
__global__ __launch_bounds__(32 * (DOUT / 16))
void wmma_gemm_kernel(const float* __restrict__ X, const float* __restrict__ W,
                      float* __restrict__ Y) {
  extern __shared__ float lds[];  // 16 x (DIN+1), +1 pad kills bank conflicts
  const int lane = threadIdx.x;          // 0..31 (wave32)
  const int wave = threadIdx.y;          // 0..DOUT/16-1
  const int m0 = blockIdx.x * 16;

  // cooperative, coalesced load of the 16xDIN stripe of X
  const int tid = wave * 32 + lane;
  const int nthr = 32 * (DOUT / 16);
  for (int i = tid; i < 16 * DIN; i += nthr) {
    int r = i / DIN, c = i % DIN;
    lds[r * (DIN + 1) + c] = X[(size_t)(m0 + r) * DIN + c];
  }
  __syncthreads();

  const int n0 = wave * 16;
  const int rr = lane & 15;
  const int khalf = (lane >> 4) * 2;  // 0 for lanes 0-15, 2 for lanes 16-31

  v8f acc = {};
#pragma unroll
  for (int k = 0; k < DIN; k += 4) {
    const int kk = k + khalf;
    v2f a, b;
    a.x = lds[rr * (DIN + 1) + kk];
    a.y = lds[rr * (DIN + 1) + kk + 1];
    b.x = W[(size_t)kk * DOUT + n0 + rr];
    b.y = W[(size_t)(kk + 1) * DOUT + n0 + rr];
    acc = __builtin_amdgcn_wmma_f32_16x16x4_f32(
        /*neg_a=*/false, a, /*neg_b=*/false, b,
        /*c_mod=*/(short)0, acc, /*reuse_a=*/false, /*reuse_b=*/false);
  }

#pragma unroll
  for (int g = 0; g < 8; ++g) {
    const int row = m0 + g + ((lane >> 4) ? 8 : 0);
    Y[(size_t)row * DOUT + n0 + rr] = acc[g];
  }
}

// ---------------- aggregation (float4 / b128 granularity) ----------------

// agg[n, 4g..4g+3] = xw[n, 4g..4g+3] * dinv[n]^2  (self-loop term; also initializes agg)
__global__ void selfloop_init_kernel(const float4* __restrict__ xw, const float* __restrict__ dinv,
                                     float4* __restrict__ agg, long long total4, int dg) {
  long long stride = (long long)gridDim.x * blockDim.x;
  for (long long i = (long long)blockIdx.x * blockDim.x + threadIdx.x; i < total4; i += stride) {
    int n = (int)(i / dg);
    float w = dinv[n];
    float w2 = w * w;
    float4 v = xw[i];
    v.x *= w2; v.y *= w2; v.z *= w2; v.w *= w2;
    agg[i] = v;
  }
}

// agg[dst, 4g..4g+3] += xw[src, 4g..4g+3] * dinv[src]*dinv[dst]
// one thread per (edge, 4-feature group): b128 gather + 4 f32 atomics
__global__ void edge_agg_kernel(const int* __restrict__ src, const int* __restrict__ dst,
                                const float* __restrict__ dinv, const float4* __restrict__ xw,
                                float* __restrict__ agg, long long total4, int dg, int dout) {
  long long stride = (long long)gridDim.x * blockDim.x;
  for (long long i = (long long)blockIdx.x * blockDim.x + threadIdx.x; i < total4; i += stride) {
    long long e = i / dg;
    int g = (int)(i - e * dg);
    int s = src[e], d = dst[e];
    float w = dinv[s] * dinv[d];
    float4 v = xw[(size_t)s * dg + g];
    float* a = agg + (size_t)d * dout + 4 * g;
    atomicAdd(a + 0, v.x * w);
    atomicAdd(a + 1, v.y * w);
    atomicAdd(a + 2, v.z * w);
    atomicAdd(a + 3, v.w * w);
  }
}

// ---------------- bias + BN(eval) + ELU (float4) ----------------

__global__ void bn_elu_kernel(const float4* __restrict__ agg, const float* __restrict__ bias,
                              const float* __restrict__ g, const float* __restrict__ be,
                              const float* __restrict__ rm, const float* __restrict__ rv,
                              float4* __restrict__ out, long long total4, int dg) {
  long long stride = (long long)gridDim.x * blockDim.x;
  for (long long i = (long long)blockIdx.x * blockDim.x + threadIdx.x; i < total4; i += stride) {
    int f = (int)(i % dg) * 4;
    float4 v = agg[i];
    float r0 = (v.x + bias[f + 0] - rm[f + 0]) * rsqrtf(rv[f + 0] + BN_EPS) * g[f + 0] + be[f + 0];
    float r1 = (v.y + bias[f + 1] - rm[f + 1]) * rsqrtf(rv[f + 1] + BN_EPS) * g[f + 1] + be[f + 1];
    float r2 = (v.z + bias[f + 2] - rm[f + 2]) * rsqrtf(rv[f + 2] + BN_EPS) * g[f + 2] + be[f + 2];
    float r3 = (v.w + bias[f + 3] - rm[f + 3]) * rsqrtf(rv[f + 3] + BN_EPS) * g[f + 3] + be[f + 3];
    float4 o;
    o.x = (r0 > 0.0f) ? r0 : expm1f(r0);
    o.y = (r1 > 0.0f) ? r1 : expm1f(r1);
    o.z = (r2 > 0.0f) ? r2 : expm1f(r2);
    o.w = (r3 > 0.0f) ? r3 : expm1f(r3);
    out[i] = o;
  }
}

// ---------------- layer 4: GEMV (din=128, dout=1) + scalar aggregation ----------------

__global__ void gemv128_kernel(const float* __restrict__ H, const float* __restrict__ w,
                               float* __restrict__ y, int n) {
  int node = blockIdx.x * blockDim.y + threadIdx.y;
  if (node >= n) return;
  int lane = threadIdx.x;  // wave32: 32 lanes x float4 = 128 features
  float4 a = ((const float4*)(H + (size_t)node * 128))[lane];
  float4 b = ((const float4*)w)[lane];
  float s = a.x * b.x + a.y * b.y + a.z * b.z + a.w * b.w;
#pragma unroll
  for (int off = 16; off > 0; off >>= 1) s += __shfl_xor(s, off, 32);
  if (lane == 0) y[node] = s;
}

__global__ void selfloop1_kernel(const float* __restrict__ xw, const float* __restrict__ dinv,
                                 float* __restrict__ agg, int n) {
  int i = blockIdx.x * blockDim.x + threadIdx.x;
  if (i < n) {
    float w = dinv[i];
    agg[i] = xw[i] * w * w;
  }
}

__global__ void edge1_kernel(const int* __restrict__ src, const int* __restrict__ dst,
                             const float* __restrict__ dinv, const float* __restrict__ xw,
                             float* __restrict__ agg, int e) {
  int i = blockIdx.x * blockDim.x + threadIdx.x;
  if (i < e) {
    int s = src[i], d = dst[i];
    atomicAdd(&agg[d], xw[s] * dinv[s] * dinv[d]);
  }
}

__global__ void final_kernel(const float* __restrict__ agg, const float* __restrict__ b4,
                             float* __restrict__ out, int n) {
  int i = blockIdx.x * blockDim.x + threadIdx.x;
  if (i < n) {
    float v = agg[i] + b4[0];
    v = (v > 0.0f) ? v : expm1f(v);       // ELU
    out[i] = 1.0f / (1.0f + expf(-v));    // sigmoid
  }
}

// ---------------- host launcher ----------------

static inline int nblk(long long total, int tb) { return (int)((total + tb - 1) / tb); }

extern "C" void kernel_launch(void* const* d_in, const int* in_sizes, int n_in,
                              void* d_out, int out_size, void* d_ws, size_t ws_size,
                              hipStream_t stream) {
  const float* x  = (const float*)d_in[0];
  const int*   ei = (const int*)d_in[1];
  const float* W1 = (const float*)d_in[2];  const float* b1 = (const float*)d_in[3];
  const float* W2 = (const float*)d_in[4];  const float* b2 = (const float*)d_in[5];
  const float* W3 = (const float*)d_in[6];  const float* b3 = (const float*)d_in[7];
  const float* W4 = (const float*)d_in[8];  const float* b4 = (const float*)d_in[9];
  const float* g1 = (const float*)d_in[10]; const float* be1 = (const float*)d_in[11];
  const float* rm1 = (const float*)d_in[12]; const float* rv1 = (const float*)d_in[13];
  const float* g2 = (const float*)d_in[14]; const float* be2 = (const float*)d_in[15];
  const float* rm2 = (const float*)d_in[16]; const float* rv2 = (const float*)d_in[17];
  const float* g3 = (const float*)d_in[18]; const float* be3 = (const float*)d_in[19];
  const float* rm3 = (const float*)d_in[20]; const float* rv3 = (const float*)d_in[21];

  const int N = in_sizes[0] / 64;     // 100000 (divisible by 16)
  const int E = in_sizes[1] / 2;      // 1600000
  const int* src = ei;
  const int* dst = ei + E;

  // workspace layout (floats): dinv(N) | xw(N*128) | agg(N*128) | h(N*128)
  // all sub-buffer byte offsets are multiples of 16 -> float4 access is aligned
  float* dinv = (float*)d_ws;
  float* xw   = dinv + N;
  float* agg  = xw + (size_t)N * 128;
  float* hbuf = agg + (size_t)N * 128;

  const int TB = 256;

  // normalization coefficients (recomputed every call -> deterministic)
  deg_init_kernel<<<nblk(N, TB), TB, 0, stream>>>(dinv, N);
  deg_accum_kernel<<<nblk(E, TB), TB, 0, stream>>>(dst, dinv, E);
  dinv_kernel<<<nblk(N, TB), TB, 0, stream>>>(dinv, N);

  // ---- layer 1: 64 -> 32 ----
  {
    dim3 blk(32, 32 / 16);
    wmma_gemm_kernel<64, 32><<<N / 16, blk, 16 * 65 * sizeof(float), stream>>>(x, W1, xw);
    const int dg = 32 / 4;
    long long tn4 = (long long)N * dg, te4 = (long long)E * dg;
    selfloop_init_kernel<<<nblk(tn4, TB), TB, 0, stream>>>((const float4*)xw, dinv, (float4*)agg, tn4, dg);
    edge_agg_kernel<<<nblk(te4, TB), TB, 0, stream>>>(src, dst, dinv, (const float4*)xw, agg, te4, dg, 32);
    bn_elu_kernel<<<nblk(tn4, TB), TB, 0, stream>>>((const float4*)agg, b1, g1, be1, rm1, rv1, (float4*)hbuf, tn4, dg);
  }
  // ---- layer 2: 32 -> 64 ----
  {
    dim3 blk(32, 64 / 16);
    wmma_gemm_kernel<32, 64><<<N / 16, blk, 16 * 33 * sizeof(float), stream>>>(hbuf, W2, xw);
    const int dg = 64 / 4;
    long long tn4 = (long long)N * dg, te4 = (long long)E * dg;
    selfloop_init_kernel<<<nblk(tn4, TB), TB, 0, stream>>>((const float4*)xw, dinv, (float4*)agg, tn4, dg);
    edge_agg_kernel<<<nblk(te4, TB), TB, 0, stream>>>(src, dst, dinv, (const float4*)xw, agg, te4, dg, 64);
    bn_elu_kernel<<<nblk(tn4, TB), TB, 0, stream>>>((const float4*)agg, b2, g2, be2, rm2, rv2, (float4*)hbuf, tn4, dg);
  }
  // ---- layer 3: 64 -> 128 ----
  {
    dim3 blk(32, 128 / 16);
    wmma_gemm_kernel<64, 128><<<N / 16, blk, 16 * 65 * sizeof(float), stream>>>(hbuf, W3, xw);
    const int dg = 128 / 4;
    long long tn4 = (long long)N * dg, te4 = (long long)E * dg;
    selfloop_init_kernel<<<nblk(tn4, TB), TB, 0, stream>>>((const float4*)xw, dinv, (float4*)agg, tn4, dg);
    edge_agg_kernel<<<nblk(te4, TB), TB, 0, stream>>>(src, dst, dinv, (const float4*)xw, agg, te4, dg, 128);
    bn_elu_kernel<<<nblk(tn4, TB), TB, 0, stream>>>((const float4*)agg, b3, g3, be3, rm3, rv3, (float4*)hbuf, tn4, dg);
  }
  // ---- layer 4: 128 -> 1, then sigmoid(elu(.)) ----
  {
    dim3 blk(32, 8);  // 8 waves, one node per wave
    gemv128_kernel<<<(N + 7) / 8, blk, 0, stream>>>(hbuf, W4, xw, N);
    selfloop1_kernel<<<nblk(N, TB), TB, 0, stream>>>(xw, dinv, agg, N);
    edge1_kernel<<<nblk(E, TB), TB, 0, stream>>>(src, dst, dinv, xw, agg, E);
    final_kernel<<<nblk(N, TB), TB, 0, stream>>>(agg, b4, (float*)d_out, N);
  }
}